// Model1_11596411699487
// MI455X (gfx1250) — compile-verified
//
#include <hip/hip_runtime.h>
#include <hip/hip_bf16.h>

typedef __attribute__((ext_vector_type(2))) float v2f;
typedef __attribute__((ext_vector_type(8))) float v8f;
typedef __attribute__((ext_vector_type(4))) unsigned int u32x4;
typedef __attribute__((ext_vector_type(8))) int i32x8;
typedef __attribute__((ext_vector_type(4))) int i32x4;

#define KCHUNK 128

#if defined(__has_builtin)
#  if __has_builtin(__builtin_amdgcn_tensor_load_to_lds)
#    define HAVE_TDM 1
#  endif
#endif
#ifndef HAVE_TDM
#  define HAVE_TDM 0
#endif

#if HAVE_TDM
// ---------------------------------------------------------------------------
// Issue a TDM tensor_load_to_lds of a 2D fp32 tile:
//   fills LDS at byte address lds_off with rows of tile_d0 elements, tile_d1 rows (0 => 1D),
//   row pitch stride0 elements in global memory. Tile packs densely in LDS.
// D# bit layout per CDNA5 ISA ch.8 (group0: count/lds/global/type,
// group1: data_size + dims + strides). Groups 2/3 zero (<=2D tensor).
// ---------------------------------------------------------------------------
__device__ __forceinline__ void tdm_load_2d_to_lds(const void* gptr, unsigned lds_off,
                                                   unsigned tile_d0, unsigned tile_d1,
                                                   unsigned long long stride0) {
    unsigned long long ga = (unsigned long long)gptr;
    const unsigned td0 = tile_d0;                    // tensor_dim0 (exact fit)
    const unsigned td1 = tile_d1 ? tile_d1 : 1u;     // tensor_dim1
    u32x4 g0;
    g0[0] = 1u;                                      // count=1, user descriptor
    g0[1] = lds_off;                                 // lds_addr (bytes)
    g0[2] = (unsigned)(ga & 0xFFFFFFFFu);            // global_addr[31:0]
    g0[3] = (unsigned)((ga >> 32) & 0x01FFFFFFu)     // global_addr[56:32]
          | (2u << 30);                              // type=2 ("image")
    i32x8 g1;
    g1[0] = 0x00020000;                              // data_size=2 (4B), mask=0
    g1[1] = (int)((td0 & 0xFFFFu) << 16);            // tensor_dim0[15:0]
    g1[2] = (int)(((td0 >> 16) & 0xFFFFu) | ((td1 & 0xFFFFu) << 16));
    g1[3] = (int)(((td1 >> 16) & 0xFFFFu) | ((tile_d0 & 0xFFFFu) << 16));
    g1[4] = (int)(tile_d1 & 0xFFFFu);                // tile_dim1 (0 => 1D), tile_dim2=0
    g1[5] = (int)(stride0 & 0xFFFFFFFFu);            // tensor_dim0_stride[31:0]
    g1[6] = (int)((stride0 >> 32) & 0xFFFFu);        // stride[47:32]; dim1 stride=0
    g1[7] = 0;
    i32x4 z4 = {0, 0, 0, 0};
#if __clang_major__ >= 23
    i32x8 z8 = {0, 0, 0, 0, 0, 0, 0, 0};
    __builtin_amdgcn_tensor_load_to_lds(g0, g1, z4, z4, z8, 0);
#else
    __builtin_amdgcn_tensor_load_to_lds(g0, g1, z4, z4, 0);
#endif
}
#endif  // HAVE_TDM

// ---------------------------------------------------------------------------
// Pack weights [Cout, Ktrue] into a zero-padded WMMA column panel:
// dst[k*Npad + n] = (n < Cout && k < Ktrue) ? src[n*Ktrue + k] : 0
// ---------------------------------------------------------------------------
__global__ void pack_w_kernel(const float* __restrict__ src, float* __restrict__ dst,
                              int Ktrue, int Kpad, int Cout, int Npad) {
    int t = blockIdx.x * blockDim.x + threadIdx.x;
    int total = Kpad * Npad;
    if (t >= total) return;
    int k = t / Npad;
    int n = t - k * Npad;
    dst[t] = (n < Cout && k < Ktrue) ? src[n * Ktrue + k] : 0.0f;
}

// ---------------------------------------------------------------------------
// Implicit-GEMM 4D convolution (valid, stride 1) via V_WMMA_F32_16X16X4_F32.
// One wave owns one 16-row M tile; 8 waves / 256-thread block.
// CONTIG=true (ksz==4 layers): a lane's K-pair is adjacent in memory (kz is
// the innermost K dim and kA%4 in {0,2}) -> single b64 gather per step.
// ---------------------------------------------------------------------------
template <bool CONTIG>
__global__ __launch_bounds__(256)
void conv4d_wmma_kernel(const float* __restrict__ X, const float* __restrict__ Wp,
                        const float* __restrict__ bias, float* __restrict__ Out,
                        int Cin, int Sin, int ksz, int Cout, int Sout,
                        int Ktrue, int Kpad) {
    __shared__ float sW[KCHUNK * 16];  // staged padded-weight panel chunk
    __shared__ int   sOff[1024];       // im2col element offsets (Kpad <= 1024)

    const int tid  = threadIdx.x;
    const int lane = tid & 31;
    const int wave = tid >> 5;
    const int col  = lane & 15;
    const int half = lane >> 4;

    const int S2 = Sin * Sin, S3 = S2 * Sin, spat_in = S3 * Sin;
    const int k2 = ksz * ksz, k3 = k2 * ksz, k4 = k3 * ksz;
    const int o2 = Sout * Sout, o3 = o2 * Sout, spat_out = o3 * Sout;

    // im2col offset table (reduction order == flattened (Cin,kw,kx,ky,kz)).
    for (int k = tid; k < Kpad; k += 256) {
        int off = 0;
        if (k < Ktrue) {
            int cin = k / k4;  int r = k - cin * k4;
            int kw  = r / k3;  r -= kw * k3;
            int kx  = r / k2;  r -= kx * k2;
            int ky  = r / ksz; int kz = r - ky * ksz;
            off = cin * spat_in + kw * S3 + kx * S2 + ky * Sin + kz;
        }
        sOff[k] = off;  // padded taps read element 0 (times a zero weight)
    }

    // Per-lane output-row decomposition -> base address of its input patch.
    const int m_base = (blockIdx.x * 8 + wave) * 16;
    const int m  = m_base + col;
    const int b  = m / spat_out;
    int s        = m - b * spat_out;
    const int ow = s / o3;   s -= ow * o3;
    const int ox = s / o2;   s -= ox * o2;
    const int oy = s / Sout; const int oz = s - oy * Sout;
    const int base = ((((b * Cin) * Sin + ow) * Sin + ox) * Sin + oy) * Sin + oz;

    const int khoff = half << 1;
#if HAVE_TDM
    const unsigned sW_off = (unsigned)(unsigned long long)(const void*)&sW[0];
#endif

    v8f c = {};
    for (int kc = 0; kc < Kpad; kc += KCHUNK) {
        const int klen = (Kpad - kc < KCHUNK) ? (Kpad - kc) : KCHUNK;
        __syncthreads();
#if HAVE_TDM
        if (tid == 0) {
            // Contiguous 1D panel chunk: klen*16 floats, DMA'd by the TDM.
            tdm_load_2d_to_lds(Wp + (size_t)kc * 16, sW_off,
                               (unsigned)(klen * 16), 0u,
                               (unsigned long long)(klen * 16));
            __builtin_amdgcn_s_wait_tensorcnt(0);
        }
#else
        for (int t = tid; t < klen * 16; t += 256)
            sW[t] = Wp[kc * 16 + t];
#endif
        if (kc + KCHUNK < Kpad)
            __builtin_prefetch(&Wp[(kc + KCHUNK) * 16 + tid], 0, 1);
        __syncthreads();

#pragma unroll 4
        for (int k0 = 0; k0 < klen; k0 += 4) {
            const int kA = kc + k0 + khoff;
            v2f a, bf;
            if (CONTIG) {
                // kz innermost & kA%4 in {0,2}: the pair is adjacent -> b64.
                __builtin_memcpy(&a, &X[base + sOff[kA]], 8);
            } else {
                a.x = X[base + sOff[kA]];
                a.y = X[base + sOff[kA + 1]];
            }
            const int wrow = (k0 + khoff) * 16 + col;
            bf.x = sW[wrow];
            bf.y = sW[wrow + 16];
            c = __builtin_amdgcn_wmma_f32_16x16x4_f32(
                    false, a, false, bf, (short)0, c, false, false);
        }
    }

    // Bias + ReLU + channel-major store (only the real Cout columns).
    const int n = col;
    if (n < Cout) {
        const float bv = bias[n];
#pragma unroll
        for (int j = 0; j < 8; ++j) {
            const int mm = m_base + j + (half << 3);
            const int bb = mm / spat_out;
            const int ss = mm - bb * spat_out;
            float v = c[j] + bv;
            v = v > 0.0f ? v : 0.0f;
            Out[(bb * Cout + n) * spat_out + ss] = v;
        }
    }
}

// ---------------------------------------------------------------------------
// Dense1: F1[256,33] = relu( H[256,1280] x Dpad[1280,48] + db1 ), WMMA tiled
// over 3 N-tiles of 16. TDM stages the strided 16-of-48 column tile.
// ---------------------------------------------------------------------------
__global__ __launch_bounds__(256)
void dense1_wmma_kernel(const float* __restrict__ H, const float* __restrict__ Dp,
                        const float* __restrict__ bias, float* __restrict__ Out) {
    __shared__ float sW[KCHUNK * 16];
    const int K = 1280, Ntrue = 33, Npad = 48;

    const int tid  = threadIdx.x;
    const int lane = tid & 31;
    const int wave = tid >> 5;
    const int col  = lane & 15;
    const int half = lane >> 4;
    const int khoff = half << 1;

    const int ntile  = blockIdx.y;                  // 0..2
    const int m_base = (blockIdx.x * 8 + wave) * 16;
    const int row    = m_base + col;
#if HAVE_TDM
    const unsigned sW_off = (unsigned)(unsigned long long)(const void*)&sW[0];
#endif

    v8f c = {};
    for (int kc = 0; kc < K; kc += KCHUNK) {
        __syncthreads();
#if HAVE_TDM
        if (tid == 0) {
            // 2D tile: 16 floats per row, KCHUNK rows, row pitch 48 floats.
            tdm_load_2d_to_lds(Dp + (size_t)kc * Npad + ntile * 16, sW_off,
                               16u, (unsigned)KCHUNK, (unsigned long long)Npad);
            __builtin_amdgcn_s_wait_tensorcnt(0);
        }
#else
        for (int t = tid; t < KCHUNK * 16; t += 256) {
            const int k = t >> 4, n = t & 15;
            sW[t] = Dp[(kc + k) * Npad + ntile * 16 + n];
        }
#endif
        __syncthreads();

#pragma unroll 4
        for (int k0 = 0; k0 < KCHUNK; k0 += 4) {
            const int kA = kc + k0 + khoff;
            v2f a, bf;
            __builtin_memcpy(&a, &H[row * K + kA], 8);   // adjacent pair -> b64
            const int wrow = (k0 + khoff) * 16 + col;
            bf.x = sW[wrow];
            bf.y = sW[wrow + 16];
            c = __builtin_amdgcn_wmma_f32_16x16x4_f32(
                    false, a, false, bf, (short)0, c, false, false);
        }
    }

    const int gn = ntile * 16 + col;
    if (gn < Ntrue) {
        const float bv = bias[gn];
#pragma unroll
        for (int j = 0; j < 8; ++j) {
            const int mm = m_base + j + (half << 3);
            float v = c[j] + bv;
            v = v > 0.0f ? v : 0.0f;
            Out[mm * Ntrue + gn] = v;
        }
    }
}

// ---------------------------------------------------------------------------
// Dense2 + softmax: out[256,2]. Tiny (K=33, N=2) -> plain VALU, 1 row/thread.
// ---------------------------------------------------------------------------
__global__ void dense2_softmax_kernel(const float* __restrict__ F1,
                                      const float* __restrict__ W2,
                                      const float* __restrict__ b2,
                                      float* __restrict__ out) {
    const int row = blockIdx.x * blockDim.x + threadIdx.x;
    if (row >= 256) return;
    float l0 = b2[0], l1 = b2[1];
#pragma unroll
    for (int i = 0; i < 33; ++i) {
        const float h = F1[row * 33 + i];
        l0 += h * W2[i];
        l1 += h * W2[33 + i];
    }
    const float mx = fmaxf(l0, l1);
    const float e0 = __expf(l0 - mx);
    const float e1 = __expf(l1 - mx);
    const float inv = 1.0f / (e0 + e1);
    out[row * 2 + 0] = e0 * inv;
    out[row * 2 + 1] = e1 * inv;
}

// ---------------------------------------------------------------------------
// Host-side orchestration (graph-capture safe: launches only).
// ---------------------------------------------------------------------------
extern "C" void kernel_launch(void* const* d_in, const int* in_sizes, int n_in,
                              void* d_out, int out_size, void* d_ws, size_t ws_size,
                              hipStream_t stream) {
    (void)in_sizes; (void)n_in; (void)out_size; (void)ws_size;

    const float* x   = (const float*)d_in[0];
    const float* w1  = (const float*)d_in[1];  const float* b1 = (const float*)d_in[2];
    const float* w2  = (const float*)d_in[3];  const float* b2 = (const float*)d_in[4];
    const float* w3  = (const float*)d_in[5];  const float* b3 = (const float*)d_in[6];
    const float* w4  = (const float*)d_in[7];  const float* b4 = (const float*)d_in[8];
    const float* w5  = (const float*)d_in[9];  const float* b5 = (const float*)d_in[10];
    const float* dw1 = (const float*)d_in[11]; const float* db1 = (const float*)d_in[12];
    const float* dw2 = (const float*)d_in[13]; const float* db2 = (const float*)d_in[14];

    float* ws = (float*)d_ws;
    // Workspace layout (floats). Ping-pong the big activations:
    //   bufA: h1 (38,880,000) -> h3 -> h5 ; bufB: h2 (15,925,248) -> h4
    const size_t A_OFF = 0;
    const size_t B_OFF = 38880000;
    size_t woff = B_OFF + 15925248;
    float* Wp1 = ws + woff; woff += (size_t)256  * 16;
    float* Wp2 = ws + woff; woff += (size_t)768  * 16;
    float* Wp3 = ws + woff; woff += (size_t)768  * 16;
    float* Wp4 = ws + woff; woff += (size_t)1024 * 16;
    float* Wp5 = ws + woff; woff += (size_t)408  * 16;
    float* Dp  = ws + woff; woff += (size_t)1280 * 48;
    float* F1  = ws + woff; woff += (size_t)256  * 33;
    float* bufA = ws + A_OFF;
    float* bufB = ws + B_OFF;

    // --- pack weights into zero-padded WMMA panels ---
    pack_w_kernel<<<(256  * 16 + 255) / 256, 256, 0, stream>>>(w1,  Wp1, 256,  256,  3,  16);
    pack_w_kernel<<<(768  * 16 + 255) / 256, 256, 0, stream>>>(w2,  Wp2, 768,  768,  3,  16);
    pack_w_kernel<<<(768  * 16 + 255) / 256, 256, 0, stream>>>(w3,  Wp3, 768,  768,  4,  16);
    pack_w_kernel<<<(1024 * 16 + 255) / 256, 256, 0, stream>>>(w4,  Wp4, 1024, 1024, 5,  16);
    pack_w_kernel<<<(408  * 16 + 255) / 256, 256, 0, stream>>>(w5,  Wp5, 405,  408,  5,  16);
    pack_w_kernel<<<(1280 * 48 + 255) / 256, 256, 0, stream>>>(dw1, Dp,  1280, 1280, 33, 48);

    // --- conv stack (grid = M/128; every M is a multiple of 128) ---
    conv4d_wmma_kernel<true ><<<12960000 / 128, 256, 0, stream>>>(x,    Wp1, b1, bufA, 1, 18, 4, 3, 15, 256,  256);
    conv4d_wmma_kernel<true ><<< 5308416 / 128, 256, 0, stream>>>(bufA, Wp2, b2, bufB, 3, 15, 4, 3, 12, 768,  768);
    conv4d_wmma_kernel<true ><<< 1679616 / 128, 256, 0, stream>>>(bufB, Wp3, b3, bufA, 3, 12, 4, 4,  9, 768,  768);
    conv4d_wmma_kernel<true ><<<  331776 / 128, 256, 0, stream>>>(bufA, Wp4, b4, bufB, 4,  9, 4, 5,  6, 1024, 1024);
    conv4d_wmma_kernel<false><<<   65536 / 128, 256, 0, stream>>>(bufB, Wp5, b5, bufA, 5,  6, 3, 5,  4, 405,  408);

    // bufA now holds h5 = [256][5*4^4] = [256][1280] contiguously (NC-major
    // layout == the reference's reshape order).
    dense1_wmma_kernel<<<dim3(2, 3), 256, 0, stream>>>(bufA, Dp, db1, F1);
    dense2_softmax_kernel<<<1, 256, 0, stream>>>(F1, dw2, db2, (float*)d_out);
}